// NLGLIF5_57647051047789
// MI455X (gfx1250) — compile-verified
//
#include <hip/hip_runtime.h>
#include <stdint.h>

// GLIF5 neuron scan for MI455X (gfx1250).
// T=500 serial steps, B*N=65536 independent neurons -> 1 thread per neuron,
// all recurrent state VGPR-resident. Bandwidth-bound: ~393 MB total traffic
// @ 23.3 TB/s => ~17us floor. No matmul => no WMMA; the CDNA5 feature that
// applies is the async global->LDS copy engine: an 8-deep ring of
// global_load_async_to_lds_b32 prefetches (ASYNCcnt-ordered) hides HBM
// latency behind the serial per-step recurrence. Streaming data uses
// non-temporal cache hints so the 192MB L2 isn't churned by read-once /
// write-once traffic.

#define TSTEPS 500
#define BDIM   16
#define NDIM   4096
#define BN     (BDIM * NDIM)   // 65536
#define BLOCK  256
#define PF     8               // async prefetch depth (ring of 8 LDS buffers)
#define DT_F   0.002f

__global__ __launch_bounds__(BLOCK) void glif5_scan_kernel(
    const float* __restrict__ I_ext,
    const float* __restrict__ C_inv,
    const float* __restrict__ alpha_p,
    const float* __restrict__ beta_p,
    const float* __restrict__ gamma_p,
    const float* __restrict__ V_rest,
    const float* __restrict__ V_reset_slope,
    const float* __restrict__ V_reset_intcp,
    const float* __restrict__ V_threshold,
    const float* __restrict__ K_theta_s,
    const float* __restrict__ A_theta_s,
    const float* __restrict__ K_theta_v,
    const float* __restrict__ A_theta_v,
    const float* __restrict__ spike_len,
    const float* __restrict__ K_Ij,   // [2, N]
    const float* __restrict__ R_Ij,   // [2, N]
    const float* __restrict__ A_Ij,   // [2, N]
    float* __restrict__ S_out,        // [B, T, N]
    float* __restrict__ V_out)        // [B, T, N]
{
    __shared__ float ibuf[PF][BLOCK];   // 8 KB ring of I_ext slices

    const int tid = threadIdx.x;
    const int i   = blockIdx.x * BLOCK + tid;   // flat (b, n)
    const int b   = i >> 12;                    // / 4096
    const int n   = i & (NDIM - 1);

    // ---- per-neuron parameters, loaded once into VGPRs ----
    const float cinv = C_inv[n];
    const float al   = alpha_p[n];
    const float be   = beta_p[n];
    const float ga   = gamma_p[n];
    const float vr   = V_rest[n];
    const float vrs  = V_reset_slope[n];
    const float vri  = V_reset_intcp[n];
    const float vth  = V_threshold[n];
    const float kts  = K_theta_s[n];
    const float ats  = A_theta_s[n];
    const float ktv  = K_theta_v[n];
    const float atv  = A_theta_v[n];
    const float slen = spike_len[n];
    const float k0   = K_Ij[n],        k1 = K_Ij[NDIM + n];
    const float r0   = R_Ij[n],        r1 = R_Ij[NDIM + n];
    const float A0   = A_Ij[n],        A1 = A_Ij[NDIM + n];

    // ---- recurrent state (VGPR-resident for all 500 steps) ----
    float V   = vr;
    float a0  = 0.0f, a1 = 0.0f;
    float thv = 0.0f, ths = 0.0f, rsw = 0.0f;

    // LDS byte offset of this lane's slot in ring buffer 0 (generic->LDS is a
    // low-32-bit truncation on gfx1250); slot k is +k*BLOCK*4 bytes.
    const uint32_t ldsbase = (uint32_t)(uintptr_t)(&ibuf[0][tid]);

    // Prologue: fill the ring — issue PF async prefetches (ASYNCcnt = PF).
    #pragma unroll
    for (int k = 0; k < PF; ++k) {
        uint32_t goff = (uint32_t)(k * BN + i) * 4u;
        uint32_t lds  = ldsbase + (uint32_t)(k * BLOCK * 4);
        asm volatile("global_load_async_to_lds_b32 %0, %1, %2 th:TH_LOAD_NT"
                     :: "v"(lds), "v"(goff), "s"(I_ext) : "memory");
    }

    const uint64_t outbase = (uint64_t)b * (uint64_t)(TSTEPS * NDIM) + (uint64_t)n;

    // Main loop: loads t..t+PF-1 are in flight at the top of iteration t.
    // Async loads of one wave complete in order, so ASYNCcnt<=PF-1 means the
    // oldest (step t) has landed. Each lane reads only its own LDS slot, so
    // no workgroup barrier is needed. After consuming slot t&(PF-1), reissue
    // it for step t+PF.
    for (int t = 0; t < TSTEPS; ++t) {
        float cur;
        if (t < TSTEPS - PF) {
            asm volatile("s_wait_asynccnt 0x7" ::: "memory");
            cur = ibuf[t & (PF - 1)][tid];
            uint32_t goff = (uint32_t)((t + PF) * BN + i) * 4u;
            uint32_t lds  = ldsbase + (uint32_t)((t & (PF - 1)) * BLOCK * 4);
            asm volatile("global_load_async_to_lds_b32 %0, %1, %2 th:TH_LOAD_NT"
                         :: "v"(lds), "v"(goff), "s"(I_ext) : "memory");
        } else {
            // Tail (last PF steps): nothing new to issue; wait for all.
            asm volatile("s_wait_asynccnt 0x0" ::: "memory");
            cur = ibuf[t & (PF - 1)][tid];
        }

        // ---- GLIF5 step (matches reference op-for-op) ----
        const float dyn_thr = vth + thv + ths;
        const float dmr     = fmaxf(DT_F - rsw, 0.0f);
        rsw                 = fmaxf(rsw - DT_F, 0.0f);
        const float dV      = V - vr;
        const float R_inv   = al + be * dV + ga * dV * dV;
        const float V_      = V - dmr * dV * cinv * R_inv
                                + dmr * (cur + a0 + a1) * cinv;
        const float S       = (V_ - dyn_thr > 0.0f) ? 1.0f : 0.0f;
        const float head    = dmr * (1.0f - (dyn_thr - V) /
                                            (fmaxf(V_, dyn_thr) - V));
        const float v_reset = vr + vri + vrs * (dyn_thr - vr);
        const float Vn      = (S == 0.0f) ? V_ : v_reset;
        const float sfd     = dmr - head;

        a0  = a0 * (1.0f - sfd * k0);
        a1  = a1 * (1.0f - sfd * k1);
        thv = thv + sfd * (-ktv * thv + atv * (Vn - vr));
        ths = ths - sfd * ths * kts;
        a0  = a0 + S * (a0 * r0 + A0);
        a1  = a1 + S * (a1 * r1 + A1);
        ths = ths + S * ats;
        rsw = rsw + S * slen - head;
        V   = Vn;

        // Coalesced write-once streams: bypass-temporal so the 192MB L2 isn't
        // displaced by 262MB of output that is never re-read on device.
        __builtin_nontemporal_store(S,  &S_out[outbase + (uint64_t)t * NDIM]);
        __builtin_nontemporal_store(Vn, &V_out[outbase + (uint64_t)t * NDIM]);
    }
}

extern "C" void kernel_launch(void* const* d_in, const int* in_sizes, int n_in,
                              void* d_out, int out_size, void* d_ws, size_t ws_size,
                              hipStream_t stream) {
    const float* I_ext         = (const float*)d_in[0];
    const float* C_inv         = (const float*)d_in[1];
    const float* alpha_p       = (const float*)d_in[2];
    const float* beta_p        = (const float*)d_in[3];
    const float* gamma_p       = (const float*)d_in[4];
    const float* V_rest        = (const float*)d_in[5];
    const float* V_reset_slope = (const float*)d_in[6];
    const float* V_reset_intcp = (const float*)d_in[7];
    const float* V_threshold   = (const float*)d_in[8];
    const float* K_theta_s     = (const float*)d_in[9];
    const float* A_theta_s     = (const float*)d_in[10];
    const float* K_theta_v     = (const float*)d_in[11];
    const float* A_theta_v     = (const float*)d_in[12];
    const float* spike_len     = (const float*)d_in[13];
    const float* K_Ij          = (const float*)d_in[14];
    const float* R_Ij          = (const float*)d_in[15];
    const float* A_Ij          = (const float*)d_in[16];

    float* S_out = (float*)d_out;                                   // [B,T,N]
    float* V_out = S_out + (size_t)BDIM * (size_t)TSTEPS * (size_t)NDIM;

    dim3 grid(BN / BLOCK);   // 256 blocks
    dim3 block(BLOCK);       // 8 wave32s per block
    glif5_scan_kernel<<<grid, block, 0, stream>>>(
        I_ext, C_inv, alpha_p, beta_p, gamma_p, V_rest, V_reset_slope,
        V_reset_intcp, V_threshold, K_theta_s, A_theta_s, K_theta_v,
        A_theta_v, spike_len, K_Ij, R_Ij, A_Ij, S_out, V_out);
}